// TauLayer_47553877901526
// MI455X (gfx1250) — compile-verified
//
#include <hip/hip_runtime.h>
#include <hip/hip_bf16.h>
#include <math.h>

#define SR_F 200.0f
#define TS 302
#define NSEG 64
#define NTHREADS 320

__global__ __launch_bounds__(NTHREADS) void tau_kernel(
    const float* __restrict__ inputs,   // [4096, 64, 3]
    const float* __restrict__ commands, // [4096]
    float* __restrict__ out)            // [4096, 302]
{
    // Raw staged params + derived per-segment records (8 floats / 32B each):
    //   [0]=onset [1]=T [2]=k_ok [3]=1/T   [4]=1/ks [5]=amplitude [6]=target [7]=pad
    __shared__ float s_raw[NSEG * 3];
    __shared__ __align__(16) float s_seg[NSEG * 8];
    __shared__ float s_total;

    const int b   = blockIdx.x;
    const int tid = threadIdx.x;

    // ---- CDNA5 async global->LDS staging of the 192 parameter floats ----
    if (tid < NSEG * 3) {
        unsigned lds_off = (unsigned)(size_t)(&s_raw[tid]); // low 32 bits of generic = LDS offset
        unsigned long long ga =
            (unsigned long long)(size_t)(inputs + (size_t)b * (NSEG * 3) + tid);
        asm volatile("global_load_async_to_lds_b32 %0, %1, off"
                     :: "v"(lds_off), "v"(ga) : "memory");
    }
    asm volatile("s_wait_asynccnt 0x0" ::: "memory");
    __syncthreads();

    // ---- Derive per-segment constants (one thread per segment) ----
    if (tid < NSEG) {
        const int   i   = tid;
        const float opt = s_raw[3 * i + 0];
        const float ks  = s_raw[3 * i + 1];
        const float dur = s_raw[3 * i + 2];

        float onset = 0.0f;                       // sum of durs[0..i-1]
        for (int j = 0; j < i; ++j) onset += s_raw[3 * j + 2];

        const float T    = (dur != 0.0f) ? dur : 1e-11f;
        const float kok  = (ks >= 0.0f && ks <= 1.0f) ? 1.0f : 0.0f;
        const float prev = (i == 0) ? commands[b] : s_raw[3 * (i - 1) + 0];

        float* sg = &s_seg[i * 8];
        sg[0] = onset;
        sg[1] = T;
        sg[2] = kok;
        sg[3] = 1.0f / T;
        sg[4] = 1.0f / ks;       // ks==0 -> +inf, matching jnp 1.0/ks
        sg[5] = prev - opt;      // amplitude = targets[i] - targets[i+1]
        sg[6] = opt;             // targets[1:][i]
        sg[7] = 0.0f;
        if (i == NSEG - 1) s_total = onset + dur;
    }
    __syncthreads();

    if (tid >= TS) return;

    const float total = s_total;
    const float nbm1  = floorf(total * SR_F + 1.0f) - 1.0f;   // nb_points - 1
    const float step  = total / fmaxf(nbm1, 1.0f);
    const float t     = fminf((float)tid, nbm1) * step;

    float sum = 0.0f;
    float cnt = 0.0f;

    #pragma unroll 4
    for (int i = 0; i < NSEG; ++i) {
        const float4 a = *(const float4*)&s_seg[i * 8];   // onset, T, kok, 1/T
        const float tvec = t - a.x;
        // f = (tvec >= 0) & (tvec/T <= 1) & (0 <= ks <= 1)
        if (tvec >= 0.0f && tvec <= a.y && a.z != 0.0f) {
            const float Tdiv = tvec * a.w;
            const float tr   = 1.0f - Tdiv * Tdiv;
            if (tr > 0.0f) {
                const float4 bd = *(const float4*)&s_seg[i * 8 + 4]; // 1/ks, amp, tgt
                // pow via exp2/log2 (v_log_f32 / v_exp_f32) reproduces JAX semantics:
                //   tr==1 & 1/ks==inf -> inf*0 -> NaN (counted, not summed)
                //   1/ks finite, tr->0+ -> -inf -> exp2 -> 0
                const float pw = __builtin_amdgcn_exp2f(bd.x * __builtin_amdgcn_logf(tr));
                const float v  = fmaf(bd.y, pw, bd.z);
                cnt += (v != 0.0f) ? 1.0f : 0.0f;  // NaN != 0 -> counts
                sum += (v == v) ? v : 0.0f;        // nansum: drop NaN
            } else {
                // pos == false: pw = 0 -> traj = target (finite)
                const float v = s_seg[i * 8 + 6];
                cnt += (v != 0.0f) ? 1.0f : 0.0f;
                sum += v;
            }
        }
        // f == 0 -> traj == 0 exactly (reference: pw forced to 0, 0 * finite)
    }

    out[(size_t)b * TS + tid] = sum / ((cnt != 0.0f) ? cnt : 1.0f);
}

extern "C" void kernel_launch(void* const* d_in, const int* in_sizes, int n_in,
                              void* d_out, int out_size, void* d_ws, size_t ws_size,
                              hipStream_t stream) {
    const float* inputs   = (const float*)d_in[0];   // (4096, 64, 3) f32
    const float* commands = (const float*)d_in[1];   // (4096,)      f32
    float*       out      = (float*)d_out;           // (4096, 302)  f32
    const int batch = in_sizes[1];                   // 4096
    tau_kernel<<<dim3(batch), dim3(NTHREADS), 0, stream>>>(inputs, commands, out);
}